// RiskPolicyValueNet_23905787969632
// MI455X (gfx1250) — compile-verified
//
#include <hip/hip_runtime.h>
#include <hip/hip_bf16.h>

// ---------------------------------------------------------------------------
// MI455X (gfx1250) implementation.
// Dominant cost: adjacency (1 GiB fp32) streamed twice -> ~86us HBM floor.
// All GEMMs use V_WMMA_F32_16X16X4_F32 (fp32 A/B, 16x16 f32 accumulators).
// k_agg uses double-buffered GLOBAL_LOAD_ASYNC_TO_LDS_B128 (ASYNCcnt) with a
// bank-conflict-free swizzled adjacency tile; weight-stationary kernels keep
// weights transposed+padded in LDS so B-fragments are single ds_load_b64.
// ---------------------------------------------------------------------------

#define N_NODES   16384
#define D_NODE    64
#define D_GLOB    32
#define D_ACT     16
#define HDIM      64
#define N_ACT     65536

typedef __attribute__((ext_vector_type(2))) float v2f;
typedef __attribute__((ext_vector_type(8))) float v8f;

#if defined(__has_builtin)
#if __has_builtin(__builtin_amdgcn_global_load_async_to_lds_b128) && \
    __has_builtin(__builtin_amdgcn_s_wait_asynccnt)
#define USE_ASYNC_LDS 1
#endif
#endif
#ifndef USE_ASYNC_LDS
#define USE_ASYNC_LDS 0
#endif

// async-copy builtin takes pointers to gcc-vector int4 in AS1 / AS3
typedef int v4i_t __attribute__((vector_size(4 * sizeof(int))));
typedef __attribute__((address_space(1))) v4i_t g_v4i;
typedef __attribute__((address_space(3))) v4i_t l_v4i;

__device__ __forceinline__ g_v4i* to_g4(const void* p) {
  return (g_v4i*)(unsigned long long)p;
}
__device__ __forceinline__ l_v4i* to_l4(void* p) {
  // generic LDS address: low 32 bits are the LDS byte offset (ISA 10.2)
  return (l_v4i*)(unsigned int)(unsigned long long)p;
}

__device__ __forceinline__ v8f wmma4(v2f a, v2f b, v8f c) {
  // D = A(16x4 f32) * B(4x16 f32) + C(16x16 f32), wave32
  return __builtin_amdgcn_wmma_f32_16x16x4_f32(false, a, false, b, (short)0, c,
                                               false, false);
}

// ---------------------------------------------------------------------------
// Kernel 1: h = relu(node_features @ Wp + bp)   [16384,64]x[64,64]
// Weights transposed+padded in LDS: B-frag = 1x ds_load_b64, banks 4n+kh.
// ---------------------------------------------------------------------------
__global__ void __launch_bounds__(256)
k_proj(const float* __restrict__ X, const float* __restrict__ W,
       const float* __restrict__ bias, float* __restrict__ Y) {
  __shared__ __align__(16) float sWT[64 * 68];  // [n][k], pad 68
  for (int i = threadIdx.x; i < 64 * 64; i += 256) {
    const int k = i >> 6, n = i & 63;
    sWT[n * 68 + k] = W[i];
  }
  __syncthreads();

  const int lane = threadIdx.x & 31;
  const int wave = threadIdx.x >> 5;
  const int mh = lane & 15;         // M index (A), N index (B/C)
  const int kh = (lane >> 4) << 1;  // K sub-offset: 0 or 2
  const int row0 = (blockIdx.x * 8 + wave) * 16;

  v8f c[4] = {};
#pragma unroll
  for (int k = 0; k < 64; k += 4) {
    v2f a = *(const v2f*)(X + (size_t)(row0 + mh) * 64 + k + kh);
#pragma unroll
    for (int t = 0; t < 4; ++t) {
      const int n = t * 16 + mh;
      v2f b = *(const v2f*)(&sWT[n * 68 + k + kh]);
      c[t] = wmma4(a, b, c[t]);
    }
  }
#pragma unroll
  for (int t = 0; t < 4; ++t) {
    const int n = t * 16 + mh;
    const float bi = bias[n];
#pragma unroll
    for (int r = 0; r < 8; ++r) {
      const int m = r + ((lane >> 4) << 3);
      float v = c[t][r] + bi;
      Y[(size_t)(row0 + m) * 64 + n] = v > 0.f ? v : 0.f;
    }
  }
}

// ---------------------------------------------------------------------------
// Kernel 2: msg = adjacency @ h    [16384,16384]x[16384,64]   <-- HBM-bound
// Block = 8 waves -> 32 rows x 64 cols. Double-buffered async LDS staging:
//   sA swizzled: element (row,col) at dword ((col>>2)*32+row)*4 + (col&3)
//     -> async b128 writes are contiguous; A-frag ds_load_b64 hits banks
//        {4*(rblk+mh)+kh} : all 32 lanes distinct.
//   sH row-major [k][64]: B-frag banks = n, conflict-free.
// Per chunk: 6 async b128 per wave; s_wait_asynccnt 6 overlaps next prefetch.
// ---------------------------------------------------------------------------
__device__ __forceinline__ void agg_stage(const float* __restrict__ Adj,
                                          const float* __restrict__ Hin,
                                          size_t row0, int k0, float* sAb,
                                          float* sHb, int t) {
#if USE_ASYNC_LDS
#pragma unroll
  for (int j = 0; j < 2; ++j) {
    const int c = t + 256 * j;  // 0..511
    const int row = c >> 4, col4 = c & 15;
    const float* g = Adj + (row0 + row) * (size_t)N_NODES + k0 + col4 * 4;
    float* l = sAb + (col4 * 32 + row) * 4;
    __builtin_amdgcn_global_load_async_to_lds_b128(to_g4(g), to_l4(l), 0, 0);
  }
#pragma unroll
  for (int j = 0; j < 4; ++j) {
    const int c = t + 256 * j;  // 0..1023
    const int kr = c >> 4, col4 = c & 15;
    const float* g = Hin + (size_t)(k0 + kr) * 64 + col4 * 4;
    float* l = sHb + kr * 64 + col4 * 4;
    __builtin_amdgcn_global_load_async_to_lds_b128(to_g4(g), to_l4(l), 0, 0);
  }
#else
#pragma unroll
  for (int j = 0; j < 2; ++j) {
    const int c = t + 256 * j;
    const int row = c >> 4, col4 = c & 15;
    const float* g = Adj + (row0 + row) * (size_t)N_NODES + k0 + col4 * 4;
    *(float4*)(sAb + (col4 * 32 + row) * 4) = *(const float4*)(g);
  }
#pragma unroll
  for (int j = 0; j < 4; ++j) {
    const int c = t + 256 * j;
    const int kr = c >> 4, col4 = c & 15;
    const float* g = Hin + (size_t)(k0 + kr) * 64 + col4 * 4;
    *(float4*)(sHb + kr * 64 + col4 * 4) = *(const float4*)(g);
  }
#endif
}

__global__ void __launch_bounds__(256)
k_agg(const float* __restrict__ Adj, const float* __restrict__ Hin,
      float* __restrict__ Msg) {
  __shared__ __align__(16) float sA[2][32 * 64];  // 2 x 8KB, swizzled
  __shared__ __align__(16) float sH[2][64 * 64];  // 2 x 16KB, row-major

  const int t = threadIdx.x;
  const int lane = t & 31;
  const int wave = t >> 5;
  const int mh = lane & 15;
  const int kh = (lane >> 4) << 1;
  const int rblk = (wave >> 2) * 16;    // 0 or 16 within 32-row block
  const int n = (wave & 3) * 16 + mh;   // output column
  const size_t row0 = (size_t)blockIdx.x * 32;

  agg_stage(Adj, Hin, row0, 0, sA[0], sH[0], t);

  v8f c = {};
  for (int i = 0; i < 256; ++i) {
    const int cur = i & 1;
    // prefetch next chunk into the other buffer ((i+1)&255 wraps harmlessly
    // on the last iteration so the asynccnt immediate stays constant)
    agg_stage(Adj, Hin, row0, ((i + 1) & 255) * 64, sA[cur ^ 1], sH[cur ^ 1],
              t);
#if USE_ASYNC_LDS
    __builtin_amdgcn_s_wait_asynccnt(6);  // chunk i complete; i+1 in flight
#endif
    __syncthreads();
    const float* sAb = sA[cur];
    const float* sHb = sH[cur];
#pragma unroll
    for (int kk = 0; kk < 64; kk += 4) {
      v2f a = *(const v2f*)(&sAb[(kk >> 2) * 128 + (rblk + mh) * 4 + kh]);
      v2f b;
      b.x = sHb[(kk + kh) * 64 + n];
      b.y = sHb[(kk + kh + 1) * 64 + n];
      c = wmma4(a, b, c);
    }
    __syncthreads();
  }
#pragma unroll
  for (int r = 0; r < 8; ++r) {
    const int m = rblk + r + ((lane >> 4) << 3);
    Msg[(row0 + m) * 64 + n] = c[r];
  }
}

// ---------------------------------------------------------------------------
// Kernel 3: hout = relu(hin @ Ws + msg @ Wn + bs + bn)
// ---------------------------------------------------------------------------
__global__ void __launch_bounds__(256)
k_combine(const float* __restrict__ Hin, const float* __restrict__ Msg,
          const float* __restrict__ Ws, const float* __restrict__ Wn,
          const float* __restrict__ bs, const float* __restrict__ bn,
          float* __restrict__ Hout) {
  __shared__ __align__(16) float sWsT[64 * 68];
  __shared__ __align__(16) float sWnT[64 * 68];
  for (int i = threadIdx.x; i < 64 * 64; i += 256) {
    const int k = i >> 6, n = i & 63;
    sWsT[n * 68 + k] = Ws[i];
    sWnT[n * 68 + k] = Wn[i];
  }
  __syncthreads();

  const int lane = threadIdx.x & 31;
  const int wave = threadIdx.x >> 5;
  const int mh = lane & 15;
  const int kh = (lane >> 4) << 1;
  const int row0 = (blockIdx.x * 8 + wave) * 16;

  v8f c[4] = {};
#pragma unroll
  for (int k = 0; k < 64; k += 4) {
    v2f a1 = *(const v2f*)(Hin + (size_t)(row0 + mh) * 64 + k + kh);
    v2f a2 = *(const v2f*)(Msg + (size_t)(row0 + mh) * 64 + k + kh);
#pragma unroll
    for (int t = 0; t < 4; ++t) {
      const int n = t * 16 + mh;
      v2f b1 = *(const v2f*)(&sWsT[n * 68 + k + kh]);
      v2f b2 = *(const v2f*)(&sWnT[n * 68 + k + kh]);
      c[t] = wmma4(a1, b1, c[t]);
      c[t] = wmma4(a2, b2, c[t]);
    }
  }
#pragma unroll
  for (int t = 0; t < 4; ++t) {
    const int n = t * 16 + mh;
    const float bi = bs[n] + bn[n];
#pragma unroll
    for (int r = 0; r < 8; ++r) {
      const int m = r + ((lane >> 4) << 3);
      float v = c[t][r] + bi;
      Hout[(size_t)(row0 + m) * 64 + n] = v > 0.f ? v : 0.f;
    }
  }
}

// ---------------------------------------------------------------------------
// Kernel 4: per-block column partial sums of h (for graph mean)
// ---------------------------------------------------------------------------
__global__ void __launch_bounds__(256)
k_rowsum(const float* __restrict__ H, float* __restrict__ Part) {
  __shared__ float red[256];
  const int col = threadIdx.x & 63;
  const int sub = threadIdx.x >> 6;  // 0..3
  const size_t base = (size_t)blockIdx.x * 256;
  float s = 0.f;
  for (int i = 0; i < 64; ++i)
    s += H[(base + sub + 4 * i) * 64 + col];
  red[threadIdx.x] = s;
  __syncthreads();
  if (sub == 0)
    Part[blockIdx.x * 64 + col] =
        red[col] + red[64 + col] + red[128 + col] + red[192 + col];
}

// ---------------------------------------------------------------------------
// Kernel 5: context / value head + precompute ctx@Wa1 (action-invariant)
// ---------------------------------------------------------------------------
__global__ void __launch_bounds__(64)
k_context(const float* __restrict__ Part, const float* __restrict__ Glob,
          const float* __restrict__ Wc, const float* __restrict__ bc,
          const float* __restrict__ Wv1, const float* __restrict__ bv1,
          const float* __restrict__ Wv2, const float* __restrict__ bv2,
          const float* __restrict__ Wa1, const float* __restrict__ ba1,
          float* __restrict__ CtxA1, float* __restrict__ ValueOut) {
  __shared__ float ge[64], gl[32], ctx[64], hv[64];
  const int j = threadIdx.x;
  float s = 0.f;
  for (int b2 = 0; b2 < 64; ++b2) s += Part[b2 * 64 + j];
  ge[j] = s * (1.0f / (float)N_NODES);
  if (j < 32) gl[j] = Glob[j];
  __syncthreads();

  float a = bc[j];
  for (int i = 0; i < 64; ++i) a += ge[i] * Wc[i * 64 + j];
  for (int i = 0; i < 32; ++i) a += gl[i] * Wc[(64 + i) * 64 + j];
  ctx[j] = a > 0.f ? a : 0.f;
  __syncthreads();

  float v = bv1[j];
  float ca = ba1[j];
  for (int i = 0; i < 64; ++i) {
    v += ctx[i] * Wv1[i * 64 + j];
    ca += ctx[i] * Wa1[i * 64 + j];  // Wa1 rows 0..63 = context slice
  }
  hv[j] = v > 0.f ? v : 0.f;
  CtxA1[j] = ca;
  __syncthreads();

  if (j == 0) {
    float val = bv2[0];
    for (int i = 0; i < 64; ++i) val += hv[i] * Wv2[i];
    ValueOut[0] = val;
  }
}

// ---------------------------------------------------------------------------
// Kernel 6: action head. One wave -> 16 actions.
// hidden = relu(ctxA1 + from@Wa1[64:128] + to@Wa1[128:192] + act@Wa1[192:208])
// Wa1 slice transposed+padded (pad 148 -> banks 20n+kh, all distinct).
// ---------------------------------------------------------------------------
__global__ void __launch_bounds__(256)
k_action(const float* __restrict__ H, const int* __restrict__ FromI,
         const int* __restrict__ ToI, const float* __restrict__ ActF,
         const float* __restrict__ Wa1, const float* __restrict__ CtxA1,
         const float* __restrict__ Wa2, const float* __restrict__ ba2,
         float* __restrict__ Logits) {
  __shared__ __align__(16) float sWT[64 * 148];  // [n][k], k=0..143
  __shared__ float sCtx[64], sW2[64];
  for (int i = threadIdx.x; i < 144 * 64; i += 256) {
    const int k = i >> 6, n = i & 63;
    sWT[n * 148 + k] = Wa1[64 * 64 + i];  // rows 64..207 of Wa1
  }
  if (threadIdx.x < 64) {
    sCtx[threadIdx.x] = CtxA1[threadIdx.x];
    sW2[threadIdx.x] = Wa2[threadIdx.x];
  }
  __syncthreads();

  const int lane = threadIdx.x & 31;
  const int wave = threadIdx.x >> 5;
  const int mh = lane & 15;
  const int kh = (lane >> 4) << 1;
  const int a0 = (blockIdx.x * 8 + wave) * 16;

  const int fi = FromI[a0 + mh];
  const int ti = ToI[a0 + mh];
  const float* pF = H + (size_t)(fi < 0 ? 0 : fi) * 64;
  const float* pT = H + (size_t)(ti < 0 ? 0 : ti) * 64;
  const float fm = fi < 0 ? 0.f : 1.f;  // -1 -> zero embedding
  const float tm = ti < 0 ? 0.f : 1.f;

  v8f c[4] = {};
#pragma unroll
  for (int k = 0; k < 64; k += 4) {  // from-embedding term
    v2f a = *(const v2f*)(pF + k + kh);
    a.x *= fm; a.y *= fm;
#pragma unroll
    for (int t = 0; t < 4; ++t) {
      const int n = t * 16 + mh;
      v2f b = *(const v2f*)(&sWT[n * 148 + k + kh]);
      c[t] = wmma4(a, b, c[t]);
    }
  }
#pragma unroll
  for (int k = 0; k < 64; k += 4) {  // to-embedding term
    v2f a = *(const v2f*)(pT + k + kh);
    a.x *= tm; a.y *= tm;
#pragma unroll
    for (int t = 0; t < 4; ++t) {
      const int n = t * 16 + mh;
      v2f b = *(const v2f*)(&sWT[n * 148 + 64 + k + kh]);
      c[t] = wmma4(a, b, c[t]);
    }
  }
#pragma unroll
  for (int k = 0; k < 16; k += 4) {  // action-feature term
    v2f a = *(const v2f*)(ActF + (size_t)(a0 + mh) * 16 + k + kh);
#pragma unroll
    for (int t = 0; t < 4; ++t) {
      const int n = t * 16 + mh;
      v2f b = *(const v2f*)(&sWT[n * 148 + 128 + k + kh]);
      c[t] = wmma4(a, b, c[t]);
    }
  }

  float sr[8];
#pragma unroll
  for (int r = 0; r < 8; ++r) sr[r] = 0.f;
#pragma unroll
  for (int t = 0; t < 4; ++t) {
    const int n = t * 16 + mh;
    const float cb = sCtx[n];
    const float w2 = sW2[n];
#pragma unroll
    for (int r = 0; r < 8; ++r) {
      float hvv = c[t][r] + cb;
      hvv = hvv > 0.f ? hvv : 0.f;
      sr[r] += hvv * w2;
    }
  }
  // reduce over the 16 lanes of each half-wave (wave32 xor shuffles)
#pragma unroll
  for (int r = 0; r < 8; ++r) {
    sr[r] += __shfl_xor(sr[r], 1, 32);
    sr[r] += __shfl_xor(sr[r], 2, 32);
    sr[r] += __shfl_xor(sr[r], 4, 32);
    sr[r] += __shfl_xor(sr[r], 8, 32);
  }
  if (mh == 0) {
    const float bb = ba2[0];
    const int mbase = (lane >> 4) << 3;
#pragma unroll
    for (int r = 0; r < 8; ++r)
      Logits[a0 + mbase + r] = sr[r] + bb;
  }
}

// ---------------------------------------------------------------------------
extern "C" void kernel_launch(void* const* d_in, const int* in_sizes, int n_in,
                              void* d_out, int out_size, void* d_ws,
                              size_t ws_size, hipStream_t stream) {
  const float* node = (const float*)d_in[0];
  const float* adj  = (const float*)d_in[1];
  const float* glob = (const float*)d_in[2];
  const float* actf = (const float*)d_in[3];
  const int* fromI  = (const int*)d_in[4];
  const int* toI    = (const int*)d_in[5];
  const float* Wp  = (const float*)d_in[6];
  const float* bp  = (const float*)d_in[7];
  const float* Ws  = (const float*)d_in[8];   // [2,64,64]
  const float* bs  = (const float*)d_in[9];   // [2,64]
  const float* Wn  = (const float*)d_in[10];  // [2,64,64]
  const float* bn  = (const float*)d_in[11];  // [2,64]
  const float* Wc  = (const float*)d_in[12];
  const float* bc  = (const float*)d_in[13];
  const float* Wa1 = (const float*)d_in[14];
  const float* ba1 = (const float*)d_in[15];
  const float* Wa2 = (const float*)d_in[16];
  const float* ba2 = (const float*)d_in[17];
  const float* Wv1 = (const float*)d_in[18];
  const float* bv1 = (const float*)d_in[19];
  const float* Wv2 = (const float*)d_in[20];
  const float* bv2 = (const float*)d_in[21];

  float* out = (float*)d_out;            // [0..65535] logits, [65536] value
  float* ws = (float*)d_ws;
  float* hA    = ws;                     // 16384*64
  float* hB    = ws + 1048576;           // 16384*64
  float* msg   = ws + 2097152;           // 16384*64
  float* part  = ws + 3145728;           // 64*64
  float* ctxa1 = ws + 3145728 + 4096;    // 64

  k_proj<<<128, 256, 0, stream>>>(node, Wp, bp, hA);
  // layer 0
  k_agg<<<512, 256, 0, stream>>>(adj, hA, msg);
  k_combine<<<128, 256, 0, stream>>>(hA, msg, Ws, Wn, bs, bn, hB);
  // layer 1
  k_agg<<<512, 256, 0, stream>>>(adj, hB, msg);
  k_combine<<<128, 256, 0, stream>>>(hB, msg, Ws + 4096, Wn + 4096, bs + 64,
                                     bn + 64, hA);
  // graph mean -> context -> value (+ ctx@Wa1 hoisted out of action loop)
  k_rowsum<<<64, 256, 0, stream>>>(hA, part);
  k_context<<<1, 64, 0, stream>>>(part, glob, Wc, bc, Wv1, bv1, Wv2, bv2, Wa1,
                                  ba1, ctxa1, out + N_ACT);
  // action head
  k_action<<<512, 256, 0, stream>>>(hA, fromI, toI, actf, Wa1, ctxa1, Wa2, ba2,
                                    out);
}